// TBSyntaxParser_56281251446796
// MI455X (gfx1250) — compile-verified
//
#include <hip/hip_runtime.h>

typedef float v2f __attribute__((ext_vector_type(2)));
typedef float v8f __attribute__((ext_vector_type(8)));

#define HID   50
#define LBUF  128
#define KDIM  300                 // 6*H concatenated feature vector
#define WAVES_PER_BLOCK 2
#define THREADS (WAVES_PER_BLOCK * 32)

// One wave32 processes a tile of 16 parser states:
//   A (16x4 f32)  = X[state, k0..k0+3]        gathered features
//   B (4x16 f32)  = W^T[k0..k0+3, out(0..2)]  zero-padded to 16 cols
//   C/D (16x16)   = running f32 accumulator
__global__ __launch_bounds__(THREADS) void tb_parser_wmma(
    const float* __restrict__ buf,    // [B, L, H]
    const float* __restrict__ W,      // [3, 300]
    const float* __restrict__ bias,   // [3]
    const float* __restrict__ legal,  // [B, 3]
    const int*   __restrict__ bidx,   // [B]
    const int*   __restrict__ sidx,   // [B, 3]
    float*       __restrict__ out,    // [B, 3]
    int nstates)
{
    __shared__ float sX[WAVES_PER_BLOCK][16][KDIM];  // 38.4 KB
    __shared__ float sW[16][KDIM];                   // 19.2 KB (rows 3..15 = 0)

    const int tid  = threadIdx.x;
    const int wave = tid >> 5;
    const int lane = tid & 31;

    // Stage zero-padded W^T source: 16*300 elems, exact multiple of THREADS.
    for (int e = tid; e < 16 * KDIM; e += THREADS) {
        const int n = e / KDIM;
        const int k = e - n * KDIM;
        sW[n][k] = (n < 3) ? W[n * KDIM + k] : 0.0f;
    }

    // Gather this wave's 16 states into LDS: 4800 elems / 32 lanes = 150 iters,
    // no divergence (EXEC stays all-ones for the WMMA loop below).
    const int waveBase = blockIdx.x * (WAVES_PER_BLOCK * 16) + wave * 16;
    for (int e = lane; e < 16 * KDIM; e += 32) {
        const int s = e / KDIM;
        const int k = e - s * KDIM;
        int state = waveBase + s;
        state = state < nstates ? state : nstates - 1;   // clamp (tail safety)
        int row, col;
        if (k < 3 * HID) {
            const int r = k / HID;
            row = bidx[state] + r;
            col = k - r * HID;
        } else {
            const int kk = k - 3 * HID;
            const int r  = kk / HID;
            row = sidx[state * 3 + r];
            col = kk - r * HID;
        }
        sX[wave][s][k] = buf[(size_t)state * (LBUF * HID) + (size_t)row * HID + col];
    }
    __syncthreads();

    // ISA A-layout (16x4 f32): lanes 0-15 -> M=lane, VGPR0/1 = K0/K1;
    //                          lanes 16-31 -> M=lane-16, VGPR0/1 = K2/K3.
    // B-layout mirrors with N in place of M.
    const int half = lane >> 4;      // 0: K pair {0,1}, 1: K pair {2,3}
    const int mn   = lane & 15;      // A row (state) / B column (output)
    const float* __restrict__ xr = &sX[wave][mn][0];
    const float* __restrict__ wr = &sW[mn][0];

    v8f acc = {0.f, 0.f, 0.f, 0.f, 0.f, 0.f, 0.f, 0.f};
    #pragma unroll 5
    for (int k0 = 0; k0 < KDIM; k0 += 4) {
        const int ka = k0 + half * 2;
        v2f a, b;
        a.x = xr[ka];     a.y = xr[ka + 1];
        b.x = wr[ka];     b.y = wr[ka + 1];
        // D = A*B + C  (f32 matrix pipe; args: neg_a, A, neg_b, B, c_mod, C, reuse_a, reuse_b)
        acc = __builtin_amdgcn_wmma_f32_16x16x4_f32(
            false, a, false, b, (short)0, acc, false, false);
    }

    // D layout: lane holds column n = lane&15; VGPR v holds row m = v + 8*(lane>>4).
    if (mn < 3) {
        const float bn = bias[mn];
        #pragma unroll
        for (int v = 0; v < 8; ++v) {
            const int state = waveBase + v + 8 * half;
            if (state < nstates) {
                float r = acc[v] + bn;
                r = fminf(r, 10.0f);
                r = fmaxf(r, -1.0e6f);
                r = __expf(r);
                out[state * 3 + mn] = r * legal[state * 3 + mn];
            }
        }
    }
}

extern "C" void kernel_launch(void* const* d_in, const int* in_sizes, int n_in,
                              void* d_out, int out_size, void* d_ws, size_t ws_size,
                              hipStream_t stream) {
    const float* buf   = (const float*)d_in[0];
    const float* W     = (const float*)d_in[1];
    const float* bias  = (const float*)d_in[2];
    const float* legal = (const float*)d_in[3];
    const int*   bidx  = (const int*)d_in[4];
    const int*   sidx  = (const int*)d_in[5];
    float*       out   = (float*)d_out;

    const int nstates = in_sizes[4];                       // len(buffer_index) == B
    const int statesPerBlock = WAVES_PER_BLOCK * 16;
    const int nblocks = (nstates + statesPerBlock - 1) / statesPerBlock;

    tb_parser_wmma<<<dim3(nblocks), dim3(THREADS), 0, stream>>>(
        buf, W, bias, legal, bidx, sidx, out, nstates);
}